// ClusterAttention_New_14620068675737
// MI455X (gfx1250) — compile-verified
//
#include <hip/hip_runtime.h>
#include <hip/hip_bf16.h>
#include <math.h>

#define DIMC   256
#define KDIM   256          // all GEMMs here have K == 256
#define HEADS  8
#define CH     32
#define MMEM   48
#define TBLSZ  2401
#define POSF   5
#define LDSP   (KDIM + 2)   // padded LDS row stride (floats), keeps b64 loads conflict-free

typedef __attribute__((ext_vector_type(2))) float v2f;
typedef __attribute__((ext_vector_type(8))) float v8f;

// ---------------------------------------------------------------------------
// Wave32 reduction helpers
// ---------------------------------------------------------------------------
__device__ __forceinline__ float wave_sum(float v) {
#pragma unroll
    for (int off = 16; off > 0; off >>= 1) v += __shfl_xor(v, off, 32);
    return v;
}
__device__ __forceinline__ float wave_max(float v) {
#pragma unroll
    for (int off = 16; off > 0; off >>= 1) v = fmaxf(v, __shfl_xor(v, off, 32));
    return v;
}

// ---------------------------------------------------------------------------
// FP32 GEMM: Cout[M x Nt] = (A[M x KDIM] * W[KDIM x Nt] + bias) * alpha
//  - Block: 256 threads = 8 waves; one 16-row A tile per block staged in LDS.
//  - Each wave owns TWO adjacent 16x16 N-tiles -> each A fragment feeds 2 WMMAs.
//  - Block covers 16 x 256 of output: grid = (M/16, Nt/256).
// ---------------------------------------------------------------------------
__global__ __launch_bounds__(256) void gemm_f32_wmma(
    const float* __restrict__ A, const float* __restrict__ W,
    const float* __restrict__ bias, float* __restrict__ Cout,
    int Nt, float alpha)
{
    __shared__ float As[16 * LDSP];

    const int tid  = threadIdx.x;
    const int lane = tid & 31;
    const int wave = tid >> 5;
    const int row0 = blockIdx.x * 16;

    // Cooperative, coalesced stage of the 16 x 256 A tile into LDS.
#pragma unroll
    for (int i = 0; i < 16; ++i)
        As[i * LDSP + tid] = A[(size_t)(row0 + i) * KDIM + tid];
    __syncthreads();

    const int l15 = lane & 15;
    const int hi  = lane >> 4;                         // lane half: 0 or 1
    const int col0 = blockIdx.y * 256 + wave * 32;     // first of two 16-col tiles

    const float* W0 = W + col0 + l15;
    const float* W1 = W + col0 + 16 + l15;
    const float* Asrow = As + l15 * LDSP + 2 * hi;

    v8f acc0 = {}, acc1 = {};
#pragma unroll 4
    for (int k = 0; k < KDIM; k += 4) {
        const v2f a = *(const v2f*)(Asrow + k);        // ds_load_b64, shared A
        const int krow = k + 2 * hi;
        v2f b0, b1;
        b0.x = W0[(size_t)krow * Nt];
        b0.y = W0[(size_t)(krow + 1) * Nt];
        b1.x = W1[(size_t)krow * Nt];
        b1.y = W1[(size_t)(krow + 1) * Nt];
        acc0 = __builtin_amdgcn_wmma_f32_16x16x4_f32(
            false, a, false, b0, (short)0, acc0, false, false);
        acc1 = __builtin_amdgcn_wmma_f32_16x16x4_f32(
            false, a, false, b1, (short)0, acc1, false, false);
    }

    const float bv0 = bias[col0 + l15];
    const float bv1 = bias[col0 + 16 + l15];
#pragma unroll
    for (int r = 0; r < 8; ++r) {
        const int row = row0 + r + 8 * hi;             // f32 C/D VGPR layout
        Cout[(size_t)row * Nt + col0 + l15]      = (acc0[r] + bv0) * alpha;
        Cout[(size_t)row * Nt + col0 + 16 + l15] = (acc1[r] + bv1) * alpha;
    }
}

// ---------------------------------------------------------------------------
// pe_table[t,h] = sum_p pre_table[t,p] * Wpe[p,h] + bpe[h]
// ---------------------------------------------------------------------------
__global__ void pe_kernel(const float* __restrict__ pre,
                          const float* __restrict__ Wpe,
                          const float* __restrict__ bpe,
                          float* __restrict__ tab, int total)
{
    int i = blockIdx.x * blockDim.x + threadIdx.x;
    if (i >= total) return;
    int t = i / HEADS, h = i - t * HEADS;
    float s = bpe[h];
#pragma unroll
    for (int p = 0; p < POSF; ++p) s += pre[t * POSF + p] * Wpe[p * HEADS + h];
    tab[i] = s;
}

// ---------------------------------------------------------------------------
// Gather attention: one block per (b,n); wave h handles head h, lane = channel.
//   kv row layout: per head h -> k at [h*64 + c], v at [h*64 + 32 + c]
//   q is pre-scaled by Ch^-0.5 (folded into the q GEMM epilogue).
// ---------------------------------------------------------------------------
__global__ __launch_bounds__(256) void attn_kernel(
    const float* __restrict__ q, const float* __restrict__ kv,
    const float* __restrict__ pe_tab,
    const int* __restrict__ member_idx, const int* __restrict__ cluster_mask,
    const int* __restrict__ pe_idx,
    const float* __restrict__ blank_k, const float* __restrict__ blank_v,
    float* __restrict__ outbuf, int N)
{
    __shared__ float sc[HEADS][52];                    // 49 used, padded

    const int lane = threadIdx.x & 31;
    const int h    = threadIdx.x >> 5;
    const int n    = blockIdx.x;
    const int b    = blockIdx.y;
    const size_t bn = (size_t)b * N + n;

    const float qv = q[bn * DIMC + h * CH + lane];     // already scaled

    const int* mrow = member_idx   + bn * MMEM;
    const int* crow = cluster_mask + bn * MMEM;
    const int* prow = pe_idx       + bn * MMEM;
    const size_t kvbase = (size_t)b * N * (2 * DIMC);

    // ---- scores: 48 gathered dots of length 32 + blank score ----
    for (int m = 0; m < MMEM; ++m) {
        const int mi = mrow[m];
        const float kvl = kv[kvbase + (size_t)mi * (2 * DIMC) + h * (2 * CH) + lane];
        const float s = wave_sum(qv * kvl);
        if (lane == 0) {
            const float pb = pe_tab[(size_t)prow[m] * HEADS + h];
            sc[h][m] = (crow[m] > 0) ? (s + pb) : -1.0e9f;
        }
    }
    {
        const float bs = wave_sum(qv * blank_k[h * CH + lane]);
        if (lane == 0) sc[h][MMEM] = bs;
    }
    __syncthreads();

    // ---- softmax over 49 entries: lane covers m=lane and m=32+lane ----
    const float e0 = sc[h][lane];
    const float e1 = (lane < 17) ? sc[h][32 + lane] : -INFINITY;
    const float mx = wave_max(fmaxf(e0, e1));
    const float p0 = __expf(e0 - mx);
    const float p1 = (lane < 17) ? __expf(e1 - mx) : 0.0f;
    const float inv = 1.0f / wave_sum(p0 + p1);
    sc[h][lane] = p0;
    if (lane < 17) sc[h][32 + lane] = p1;
    __syncthreads();

    // ---- weighted sum of gathered v rows + blank value ----
    float acc = 0.0f;
    for (int m = 0; m < MMEM; ++m) {
        const int mi = mrow[m];
        const float vvl = kv[kvbase + (size_t)mi * (2 * DIMC) + h * (2 * CH) + CH + lane];
        acc += sc[h][m] * vvl;
    }
    acc += sc[h][MMEM] * blank_v[h * CH + lane];
    outbuf[bn * DIMC + h * CH + lane] = acc * inv;
}

// ---------------------------------------------------------------------------
extern "C" void kernel_launch(void* const* d_in, const int* in_sizes, int n_in,
                              void* d_out, int out_size, void* d_ws, size_t ws_size,
                              hipStream_t stream)
{
    (void)in_sizes; (void)n_in; (void)out_size; (void)ws_size;

    const float* feat         = (const float*)d_in[0];
    const int*   member_idx   = (const int*)d_in[1];
    const int*   cluster_mask = (const int*)d_in[2];
    const int*   pe_idx       = (const int*)d_in[3];
    /* d_in[4] = global_attn (unused) */
    const float* pre_table    = (const float*)d_in[5];
    const float* Wq           = (const float*)d_in[6];
    const float* bq           = (const float*)d_in[7];
    const float* Wkv          = (const float*)d_in[8];
    const float* bkv          = (const float*)d_in[9];
    const float* blank_k      = (const float*)d_in[10];
    const float* blank_v      = (const float*)d_in[11];
    const float* Wpe          = (const float*)d_in[12];
    const float* bpe          = (const float*)d_in[13];
    const float* Wproj        = (const float*)d_in[14];
    const float* bproj        = (const float*)d_in[15];

    const int B = 2, N = 4096;
    const int BN = B * N;                               // 8192 rows
    const float scale = 0.1767766952966369f;            // Ch^-0.5, Ch=32

    float* ws    = (float*)d_ws;
    float* qbuf  = ws;                                  // BN*256
    float* kvbuf = qbuf  + (size_t)BN * DIMC;           // BN*512
    float* abuf  = kvbuf + (size_t)BN * 2 * DIMC;       // BN*256
    float* tab   = abuf  + (size_t)BN * DIMC;           // TBL*H

    const dim3 blk(256, 1, 1);

    // q = (feat @ Wq + bq) * scale   (8192 x 256 x 256)
    gemm_f32_wmma<<<dim3(BN / 16, 1), blk, 0, stream>>>(
        feat, Wq, bq, qbuf, DIMC, scale);
    // kv = feat @ Wkv + bkv          (8192 x 256 x 512)
    gemm_f32_wmma<<<dim3(BN / 16, 2), blk, 0, stream>>>(
        feat, Wkv, bkv, kvbuf, 2 * DIMC, 1.0f);
    // pe_table = pre_table @ Wpe + bpe
    pe_kernel<<<(TBLSZ * HEADS + 255) / 256, blk, 0, stream>>>(
        pre_table, Wpe, bpe, tab, TBLSZ * HEADS);
    // gather attention (per (b,n) block, per-head waves)
    attn_kernel<<<dim3(N, B), blk, 0, stream>>>(
        qbuf, kvbuf, tab, member_idx, cluster_mask, pe_idx,
        blank_k, blank_v, abuf, N);
    // out = attn_out @ Wproj + bproj (8192 x 256 x 256)
    gemm_f32_wmma<<<dim3(BN / 16, 1), blk, 0, stream>>>(
        abuf, Wproj, bproj, (float*)d_out, DIMC, 1.0f);
}